// ModelNew_73315091743861
// MI455X (gfx1250) — compile-verified
//
#include <hip/hip_runtime.h>

typedef __attribute__((ext_vector_type(2))) float v2f;
typedef __attribute__((ext_vector_type(8))) float v8f;

#define NDIM 4096
#define BM 128
#define BN 128
#define BK 16
#define LDA 20                 // floats per A-tile row (16 + 4 pad): conflict-free frag reads
#define LDB2 144               // float2 per B pair-row (128 + 16 pad): disjoint half-wave banks
#define A_ELEMS (BM * LDA)             // 2560 floats (10 KB)
#define B_ELEMS ((BK / 2) * LDB2 * 2)  // 2304 floats (9 KB)
#define NTHREADS 256

#if defined(__HIP_DEVICE_COMPILE__) && \
    __has_builtin(__builtin_amdgcn_global_load_async_to_lds_b128) && \
    __has_builtin(__builtin_amdgcn_s_wait_asynccnt)
#define USE_ASYNC 1
typedef int v4i __attribute__((vector_size(16)));
typedef __attribute__((address_space(1))) v4i* as1_v4i_p;
typedef __attribute__((address_space(3))) v4i* as3_v4i_p;
#else
#define USE_ASYNC 0
#endif

// ---- A tile, masked to triu(A): keep A[i][k] iff k >= i (diagonal band) ----
__device__ __forceinline__ void load_A_masked(const float* __restrict__ A,
                                              int i0, int k0,
                                              float* __restrict__ As, int tid)
{
    #pragma unroll
    for (int r = 0; r < (BM * BK / 4) / NTHREADS; ++r) {   // 2 iters
        int f   = tid + NTHREADS * r;
        int row = f >> 2;              // BK/4 = 4 float4 per row
        int c4  = (f & 3) << 2;
        int ig  = i0 + row;
        int kg  = k0 + c4;
        float4 v = *(const float4*)(A + (size_t)ig * NDIM + kg);
        v.x = (kg + 0 >= ig) ? v.x : 0.0f;
        v.y = (kg + 1 >= ig) ? v.y : 0.0f;
        v.z = (kg + 2 >= ig) ? v.z : 0.0f;
        v.w = (kg + 3 >= ig) ? v.w : 0.0f;
        *(float4*)(As + row * LDA + c4) = v;
    }
}

// ---- A tile, pure copy (tile entirely above the diagonal) ----
__device__ __forceinline__ void load_A_plain(const float* __restrict__ A,
                                             int i0, int k0,
                                             float* __restrict__ As, int tid)
{
    #pragma unroll
    for (int r = 0; r < (BM * BK / 4) / NTHREADS; ++r) {   // 2 iters
        int f   = tid + NTHREADS * r;
        int row = f >> 2;
        int c4  = (f & 3) << 2;
        const float* g = A + (size_t)(i0 + row) * NDIM + k0 + c4;
        float*       l = As + row * LDA + c4;
#if USE_ASYNC
        // global_load_async_to_lds_b128: 16B/lane, no VGPR staging, ASYNCcnt
        __builtin_amdgcn_global_load_async_to_lds_b128(
            (as1_v4i_p)(uintptr_t)g, (as3_v4i_p)(uintptr_t)l, 0, 0);
#else
        *(float4*)l = *(const float4*)g;
#endif
    }
}

// ---- B tile staging helper: K-pair-interleave into LDS ----
__device__ __forceinline__ void store_B_pair(float* __restrict__ Bs,
                                             int rp, int c4,
                                             float4 v0, float4 v1)
{
    float4 w0 = make_float4(v0.x, v1.x, v0.y, v1.y);
    float4 w1 = make_float4(v0.z, v1.z, v0.w, v1.w);
    float* dst = Bs + rp * (LDB2 * 2) + (c4 << 1);
    *(float4*)(dst + 0) = w0;
    *(float4*)(dst + 4) = w1;
}

// ---- B tile, masked to triu(B): keep B[k][j] iff k <= j (diagonal band) ----
__device__ __forceinline__ void load_B_masked(const float* __restrict__ B,
                                              int j0, int k0,
                                              float* __restrict__ Bs, int tid)
{
    #pragma unroll
    for (int r = 0; r < ((BK / 2) * (BN / 4)) / NTHREADS; ++r) {  // 1 iter
        int f   = tid + NTHREADS * r;
        int rp  = f >> 5;              // BN/4 = 32 float4-units per pair-row
        int c4  = (f & 31) << 2;
        int kg  = k0 + 2 * rp;
        int jg  = j0 + c4;
        float4 v0 = *(const float4*)(B + (size_t)(kg + 0) * NDIM + jg);
        float4 v1 = *(const float4*)(B + (size_t)(kg + 1) * NDIM + jg);
        v0.x = (kg + 0 <= jg + 0) ? v0.x : 0.0f;
        v0.y = (kg + 0 <= jg + 1) ? v0.y : 0.0f;
        v0.z = (kg + 0 <= jg + 2) ? v0.z : 0.0f;
        v0.w = (kg + 0 <= jg + 3) ? v0.w : 0.0f;
        v1.x = (kg + 1 <= jg + 0) ? v1.x : 0.0f;
        v1.y = (kg + 1 <= jg + 1) ? v1.y : 0.0f;
        v1.z = (kg + 1 <= jg + 2) ? v1.z : 0.0f;
        v1.w = (kg + 1 <= jg + 3) ? v1.w : 0.0f;
        store_B_pair(Bs, rp, c4, v0, v1);
    }
}

// ---- B tile, unmasked copy + interleave ----
__device__ __forceinline__ void load_B_plain(const float* __restrict__ B,
                                             int j0, int k0,
                                             float* __restrict__ Bs, int tid)
{
    #pragma unroll
    for (int r = 0; r < ((BK / 2) * (BN / 4)) / NTHREADS; ++r) {  // 1 iter
        int f   = tid + NTHREADS * r;
        int rp  = f >> 5;
        int c4  = (f & 31) << 2;
        int kg  = k0 + 2 * rp;
        int jg  = j0 + c4;
        float4 v0 = *(const float4*)(B + (size_t)(kg + 0) * NDIM + jg);
        float4 v1 = *(const float4*)(B + (size_t)(kg + 1) * NDIM + jg);
        store_B_pair(Bs, rp, c4, v0, v1);
    }
}

__device__ __forceinline__ void load_tiles(const float* __restrict__ A,
                                           const float* __restrict__ B,
                                           int i0, int j0, int k0,
                                           float* __restrict__ As,
                                           float* __restrict__ Bs, int tid)
{
    // Uniform (scalar) branches: whole block takes the same path per k-tile.
    if (k0 < i0 + BM) load_A_masked(A, i0, k0, As, tid);
    else              load_A_plain (A, i0, k0, As, tid);
    if (k0 + BK > j0) load_B_masked(B, j0, k0, Bs, tid);
    else              load_B_plain (B, j0, k0, Bs, tid);
}

__device__ __forceinline__ void sync_tiles()
{
#if USE_ASYNC
    __builtin_amdgcn_s_wait_asynccnt(0);   // my async LDS writes done
#endif
    __syncthreads();                       // everyone's writes visible
}

__global__ __launch_bounds__(NTHREADS)
void triu_mm_wmma(const float* __restrict__ A,
                  const float* __restrict__ B,
                  float* __restrict__ C)
{
    __shared__ float sA[2 * A_ELEMS];
    __shared__ float sB[2 * B_ELEMS];

    const int tid = threadIdx.x;
    const int i0  = blockIdx.y * BM;   // output row block
    const int j0  = blockIdx.x * BN;   // output col block

    // Both i0 and j0 are multiples of 128: block has any i<=j iff i0 <= j0.
    if (i0 > j0) {
        #pragma unroll
        for (int r = 0; r < (BM * BN / 4) / NTHREADS; ++r) {  // 16 iters
            int f   = tid + NTHREADS * r;
            int row = f >> 5;
            int c4  = (f & 31) << 2;
            float4 z = make_float4(0.f, 0.f, 0.f, 0.f);
            *(float4*)(C + (size_t)(i0 + row) * NDIM + j0 + c4) = z;
        }
        return;
    }

    const int lane = tid & 31;
    const int wave = tid >> 5;
    const int wm   = wave & 3;    // M offset = wm*32
    const int wn   = wave >> 2;   // N offset = wn*64
    const int m16  = lane & 15;
    const int kh   = lane >> 4;   // which K-half of the frag this lane holds

    v8f acc[2][4] = {};

    const int kend = j0 + BN;     // k in [i0, j0+BN) covers all i<=k<=j in block

    load_tiles(A, B, i0, j0, i0, sA, sB, tid);
    sync_tiles();

    int cur = 0;
    for (int k0 = i0; k0 < kend; k0 += BK) {
        if (k0 + BK < kend)
            load_tiles(A, B, i0, j0, k0 + BK,
                       sA + (cur ^ 1) * A_ELEMS,
                       sB + (cur ^ 1) * B_ELEMS, tid);

        const float* As = sA + cur * A_ELEMS;
        const float* Bs = sB + cur * B_ELEMS;

        #pragma unroll
        for (int ks = 0; ks < BK; ks += 4) {
            const int p = (ks >> 1) + kh;          // interleaved pair-row
            v2f a[2], b[4];
            const float* bp = Bs + p * (LDB2 * 2) + ((wn * 64 + m16) << 1);
            #pragma unroll
            for (int u = 0; u < 4; ++u) {          // contiguous 8B: ds_load_b64
                b[u].x = bp[u * 32 + 0];
                b[u].y = bp[u * 32 + 1];
            }
            #pragma unroll
            for (int t = 0; t < 2; ++t) {
                const float* ap = As + (wm * 32 + t * 16 + m16) * LDA + ks + 2 * kh;
                a[t].x = ap[0];
                a[t].y = ap[1];
            }
            #pragma unroll
            for (int t = 0; t < 2; ++t)
                #pragma unroll
                for (int u = 0; u < 4; ++u)
                    acc[t][u] = __builtin_amdgcn_wmma_f32_16x16x4_f32(
                        false, a[t], false, b[u], (short)0, acc[t][u],
                        false, false);
        }
        sync_tiles();
        cur ^= 1;
    }

    // Store: masked inputs make everything below the diagonal exactly 0.0,
    // so no output masking needed for processed blocks.
    #pragma unroll
    for (int t = 0; t < 2; ++t) {
        #pragma unroll
        for (int u = 0; u < 4; ++u) {
            const int gi = i0 + wm * 32 + t * 16 + 8 * kh;
            const int gj = j0 + wn * 64 + u * 16 + m16;
            #pragma unroll
            for (int v = 0; v < 8; ++v) {
                C[(size_t)(gi + v) * NDIM + gj] = acc[t][u][v];
            }
        }
    }
}

extern "C" void kernel_launch(void* const* d_in, const int* in_sizes, int n_in,
                              void* d_out, int out_size, void* d_ws, size_t ws_size,
                              hipStream_t stream) {
    (void)in_sizes; (void)n_in; (void)d_ws; (void)ws_size; (void)out_size;
    const float* A = (const float*)d_in[0];
    const float* B = (const float*)d_in[1];
    float*       C = (float*)d_out;

    dim3 grid(NDIM / BN, NDIM / BM);   // (32, 32)
    dim3 block(NTHREADS);
    hipLaunchKernelGGL(triu_mm_wmma, grid, block, 0, stream, A, B, C);
}